// StructuredRandomFeatures_446676599050
// MI455X (gfx1250) — compile-verified
//
#include <hip/hip_runtime.h>
#include <hip/hip_bf16.h>
#include <stdint.h>
#include <math.h>

typedef __attribute__((ext_vector_type(2))) float v2f;
typedef __attribute__((ext_vector_type(8))) float v8f;
typedef int v4i __attribute__((vector_size(16)));

// Padded LDS layout: logical float index j -> j + 4*(j>>4)  (4-float pad per 16)
#define LDSPAD(j) ((j) + (((j) >> 4) << 2))

__device__ __forceinline__ float h16sign(int m, int k) {
  return (__builtin_popcount(m & k) & 1) ? -1.0f : 1.0f;
}

// One wave (32 lanes) per batch row. Block = 32 threads.
__global__ __launch_bounds__(32) void srf_kernel(
    const float* __restrict__ x,    // [8192, 4096]
    const float* __restrict__ D1,   // [4096] +-1
    const float* __restrict__ D2,   // [4096] +-1
    const float* __restrict__ D3,   // [4096] +-1
    const float* __restrict__ bias, // [2048]
    const int*   __restrict__ perm, // [4096]
    float* __restrict__ out)        // [8192, 2048]
{
  __shared__ __align__(16) float Abuf[5120]; // 20 KB
  __shared__ __align__(16) float Bbuf[5120]; // 20 KB

  const int lane = threadIdx.x;   // 0..31
  const int row  = blockIdx.x;    // 0..8191

  const int n  = lane & 15;       // WMMA B/D column index (N)
  const int kh = lane >> 4;       // lane half (selects K/M upper groups)

  // Column 'n' -> input-index bits {0,2,6,7}; output-index bits {3,7,0,2}
  const int n0 = n & 1, n1 = (n >> 1) & 1, n2 = (n >> 2) & 1, n3 = (n >> 3) & 1;
  const int jn = (n0 << 0) | (n1 << 2) | (n2 << 6) | (n3 << 7);
  const int iN = (n0 << 3) | (n1 << 7) | (n2 << 0) | (n3 << 2);

  // Constant H16 A-operands. A layout (16x4 f32): element (m,k') at
  // lane = m + 16*(k'>>1), vgpr = k'&1  ==>  per lane m = n, k' = p + 2*kh.
  v2f hA[4];
  const int m = n;
#pragma unroll
  for (int kc = 0; kc < 4; ++kc) {
    const int kb = 4 * kc + 2 * kh;
    hA[kc].x = h16sign(m, kb);
    hA[kc].y = h16sign(m, kb + 1);
  }

  // ---------------- Load row x[row,:] into Abuf (padded layout) -------------
  const float* xrow = x + (size_t)row * 4096u;
#if __has_builtin(__builtin_amdgcn_global_load_async_to_lds_b128)
#pragma unroll
  for (int it = 0; it < 32; ++it) {
    const int j   = (it * 32 + lane) * 4;       // 4 consecutive floats
    const int off = LDSPAD(j);                  // stays 16B-aligned
    __builtin_amdgcn_global_load_async_to_lds_b128(
        (__attribute__((address_space(1))) v4i*)(uintptr_t)(xrow + j),
        (__attribute__((address_space(3))) v4i*)(uint32_t)(uintptr_t)(&Abuf[off]),
        0, 0);
  }
#if __has_builtin(__builtin_amdgcn_s_wait_asynccnt)
  __builtin_amdgcn_s_wait_asynccnt(0);
#else
  asm volatile("s_wait_asynccnt 0" ::: "memory");
#endif
  asm volatile("" ::: "memory");
#else
#pragma unroll
  for (int it = 0; it < 32; ++it) {
    const int j = (it * 32 + lane) * 4;
    const float4 v = *(const float4*)(xrow + j);
    *(float4*)(&Abuf[LDSPAD(j)]) = v;
  }
#endif

  // ---------------- Two "fwht" passes: y = (1/4) * H16xP * (D .* v) ---------
  // Coupled bits:  a = (j3,j4,j5,j1)  -> b = (i1,i5,i9,i11)
  // Identity bits: (j0,j2,j6,j7 | j8..j11) -> (i3,i7,i0,i2 | i4,i6,i8,i10)
#pragma unroll 1
  for (int pass = 0; pass < 2; ++pass) {
    const float* src = pass ? Bbuf : Abuf;
    float*       dst = pass ? Abuf : Bbuf;
    const float* Dm  = pass ? D2 : D1;

#pragma unroll 1
    for (int t = 0; t < 16; ++t) {           // 16 tiles of 16 columns (u = t*16+n)
      v8f c = {0.0f, 0.0f, 0.0f, 0.0f, 0.0f, 0.0f, 0.0f, 0.0f};
#pragma unroll
      for (int kc = 0; kc < 4; ++kc) {       // K = 16 in chunks of 4
        const int ka = 4 * kc + 2 * kh;      // B element k' = p + 2*kh
        const int kb = ka + 1;
        // k-bits -> input index bits: k0->j3, k1->j4, k2->j5, k3->j1
        const int ja = jn | ((ka & 7) << 3) | ((ka >> 3) << 1) | (t << 8);
        const int jb = jn | ((kb & 7) << 3) | ((kb >> 3) << 1) | (t << 8);
        v2f b;
        b.x = src[LDSPAD(ja)] * Dm[ja] * 0.25f;
        b.y = src[LDSPAD(jb)] * Dm[jb] * 0.25f;
        c = __builtin_amdgcn_wmma_f32_16x16x4_f32(
            false, hA[kc], false, b, (short)0, c, false, false);
      }
      // D layout: vgpr r, lane l: M = r + 8*(l>>4), N = l&15
      // b-bits -> output bits: b0->i1, b1->i5, b2->i9, b3->i11
      // t-bits -> output bits: t0->i4, t1->i6, t2->i8, t3->i10
      const int iT = ((t & 1) << 4) | (((t >> 1) & 1) << 6) |
                     (((t >> 2) & 1) << 8) | (((t >> 3) & 1) << 10);
#pragma unroll
      for (int r = 0; r < 8; ++r) {
        const int i = iN | iT | ((r & 1) << 1) | (((r >> 1) & 1) << 5) |
                      (((r >> 2) & 1) << 9) | (kh << 11);
        dst[LDSPAD(i)] = c[r];
      }
    }
    // wave-private LDS, DS ops are in-order per wave: no barrier needed
  }

  // ---------------- out[j] = (1/32) * cos(D3[p]*y2[p] + bias[j]), p=perm[j] --
  float* orow = out + (size_t)row * 2048u;
#pragma unroll 4
  for (int it = 0; it < 64; ++it) {
    const int jo = it * 32 + lane;
    const int p  = perm[jo];
    const float v = Abuf[LDSPAD(p)] * D3[p];
    orow[jo] = 0.03125f * cosf(v + bias[jo]);
  }
}

extern "C" void kernel_launch(void* const* d_in, const int* in_sizes, int n_in,
                              void* d_out, int out_size, void* d_ws, size_t ws_size,
                              hipStream_t stream) {
  const float* x    = (const float*)d_in[0];
  const float* D1   = (const float*)d_in[1];
  const float* D2   = (const float*)d_in[2];
  const float* D3   = (const float*)d_in[3];
  const float* bias = (const float*)d_in[4];
  const int*   perm = (const int*)d_in[5];
  float* out = (float*)d_out;
  (void)in_sizes; (void)n_in; (void)out_size; (void)d_ws; (void)ws_size;

  srf_kernel<<<dim3(8192), dim3(32), 0, stream>>>(x, D1, D2, D3, bias, perm, out);
}